// GNNEdgeExtractor_53936199303380
// MI455X (gfx1250) — compile-verified
//
#include <hip/hip_runtime.h>

// GCN forward for MI455X (gfx1250, wave32).
// - GEMMs: V_WMMA_F32_16X16X4_F32; one wave owns a 16-row strip and loops all
//   F/16 column tiles reusing the A registers (independent accumulators break
//   WMMA->WMMA RAW hazards). Previous layer's bias+ReLU is fused into the
//   A-load prologue, output bias into the epilogue (FC only).
// - Edge aggregation: 16 lanes per edge -> contiguous 64B bursts + f32 atomics;
//   per-layer feature matrix (<=64MB) + edge index (16MB) fit the 192MB L2.

typedef __attribute__((ext_vector_type(2))) float v2f;
typedef __attribute__((ext_vector_type(8))) float v8f;

// ---------------- degree / normalization ----------------
__global__ void k_fill_one(float* __restrict__ deg, int n) {
  int i = blockIdx.x * blockDim.x + threadIdx.x;
  if (i < n) deg[i] = 1.0f;              // self-loop contributes 1
}

__global__ void k_count_deg(const int* __restrict__ dst, float* __restrict__ deg, int e_cnt) {
  int e = blockIdx.x * blockDim.x + threadIdx.x;
  if (e < e_cnt) atomicAdd(&deg[dst[e]], 1.0f);
}

__global__ void k_to_dinv(float* __restrict__ deg, int n) {
  int i = blockIdx.x * blockDim.x + threadIdx.x;
  if (i < n) deg[i] = 1.0f / sqrtf(deg[i]);   // deg >= 1 always
}

// ---- WMMA GEMM: Y[n x F] = act(X[n x K]) @ W[K x F] (+ bias) -------------
// PRE:  apply previous layer's epilogue on load: a = relu(a + pre_b[k])
// BIAS: add output bias (final FC only)
// One wave per 16-row strip; loops F/16 column tiles with A kept in registers.
template<int K, int F, bool PRE, bool BIAS>
__global__ void k_gemm_wmma(const float* __restrict__ X, const float* __restrict__ W,
                            const float* __restrict__ pre_b, const float* __restrict__ bias,
                            float* __restrict__ Y, int n) {
  constexpr int NTN = F / 16;
  int wave = (blockIdx.x * blockDim.x + threadIdx.x) >> 5;   // global wave id
  int lane = threadIdx.x & 31;
  if (wave >= (n >> 4)) return;        // wave-uniform exit (EXEC stays all-1s)

  int lid = lane & 15;
  int lh  = lane >> 4;                 // lane-half: picks K pair / M half
  const float* xr = X + (size_t)(wave * 16 + lid) * K;   // A row for this lane

  v8f c[NTN];
#pragma unroll
  for (int t = 0; t < NTN; ++t) c[t] = (v8f){};

#pragma unroll
  for (int k = 0; k < K; k += 4) {
    const int k0 = k + 2 * lh;
    // A 16x4: VGPR0 = K=k0, VGPR1 = K=k0+1  (M = lane%16)
    v2f a;
    a[0] = xr[k0];
    a[1] = xr[k0 + 1];
    if constexpr (PRE) {               // fused relu(x + b_prev)
      a[0] = fmaxf(a[0] + pre_b[k0], 0.0f);
      a[1] = fmaxf(a[1] + pre_b[k0 + 1], 0.0f);
    }
    const float* wr = W + (size_t)k0 * F + lid;
#pragma unroll
    for (int t = 0; t < NTN; ++t) {
      // B 4x16: VGPR0 = row K=k0, VGPR1 = row K=k0+1  (N = lane%16)
      v2f b;
      b[0] = wr[t * 16];
      b[1] = wr[t * 16 + F];
      c[t] = __builtin_amdgcn_wmma_f32_16x16x4_f32(false, a, false, b, (short)0,
                                                   c[t], false, false);
    }
  }

#pragma unroll
  for (int t = 0; t < NTN; ++t) {
    int col = t * 16 + lid;
    float bv = 0.0f;
    if constexpr (BIAS) bv = bias[col];
    // C/D: VGPR v -> M = v + 8*lh, N = col
    float* yp = Y + (size_t)(wave * 16 + 8 * lh) * F + col;
#pragma unroll
    for (int v = 0; v < 8; ++v) {
      float val = c[t][v];
      if constexpr (BIAS) val += bv;
      yp[(size_t)v * F] = val;
    }
  }
}

// ---------------- aggregation ----------------
// out[i] = xw[i] * dinv[i]^2   (self-loop term; also initializes out)
template<int F>
__global__ void k_init_self(const float* __restrict__ xw, const float* __restrict__ dinv,
                            float* __restrict__ out, int n) {
  constexpr int SH = (F == 16) ? 4 : (F == 32) ? 5 : 6;
  long long t = (long long)blockIdx.x * blockDim.x + threadIdx.x;
  if (t >= (long long)n * F) return;
  int i = (int)(t >> SH);
  float di = dinv[i];
  out[t] = xw[t] * di * di;
}

// out[dst] += xw[src] * dinv[src]*dinv[dst]; 16 lanes cooperate on one edge.
template<int F>
__global__ void k_edge_scatter(const float* __restrict__ xw, const int* __restrict__ src,
                               const int* __restrict__ dst, const float* __restrict__ dinv,
                               float* __restrict__ out, int e_cnt) {
  long long t = (long long)blockIdx.x * blockDim.x + threadIdx.x;
  int e = (int)(t >> 4);
  int lid = (int)(t & 15);
  if (e >= e_cnt) return;
  int s = src[e], d = dst[e];
  float w = dinv[s] * dinv[d];
  const float* xs = xw + (size_t)s * F;
  float* od = out + (size_t)d * F;
#pragma unroll
  for (int f = lid; f < F; f += 16) {
    atomicAdd(&od[f], xs[f] * w);
  }
}

// ---------------- host-side layer driver ----------------
template<int K, int F, bool PRE>
static void run_layer(const float* h_in, const float* W, const float* pre_b,
                      const float* dinv, const int* src, const int* dst,
                      float* xw, float* out, int n, int e, hipStream_t stream) {
  const int B = 256;
  long long gemm_threads = (long long)(n / 16) * 32;   // one wave per row strip
  k_gemm_wmma<K, F, PRE, false><<<dim3((gemm_threads + B - 1) / B), dim3(B), 0, stream>>>(
      h_in, W, pre_b, nullptr, xw, n);
  long long nf = (long long)n * F;
  k_init_self<F><<<dim3((nf + B - 1) / B), dim3(B), 0, stream>>>(xw, dinv, out, n);
  long long et = (long long)e * 16;
  k_edge_scatter<F><<<dim3((et + B - 1) / B), dim3(B), 0, stream>>>(xw, src, dst, dinv, out, e);
  // NOTE: 'out' is left RAW (pre bias+relu); the next GEMM applies relu(out+b)
  // in its A-load prologue.
}

extern "C" void kernel_launch(void* const* d_in, const int* in_sizes, int n_in,
                              void* d_out, int out_size, void* d_ws, size_t ws_size,
                              hipStream_t stream) {
  const float* x   = (const float*)d_in[0];
  const float* W1  = (const float*)d_in[1];
  const float* b1  = (const float*)d_in[2];
  const float* W2  = (const float*)d_in[3];
  const float* b2  = (const float*)d_in[4];
  const float* W3  = (const float*)d_in[5];
  const float* b3  = (const float*)d_in[6];
  const float* Wfc = (const float*)d_in[7];
  const float* bfc = (const float*)d_in[8];
  const int*   ei  = (const int*)d_in[9];

  const int n = in_sizes[0] / 8;    // F_IN = 8
  const int e = in_sizes[9] / 2;    // edge_index is (2, E)
  const int* src = ei;
  const int* dst = ei + e;

  // workspace: dinv (n floats) + two n*64 ping-pong buffers (~129MB)
  float* dinv = (float*)d_ws;
  float* bufA = dinv + n;                 // xw (gemm output)
  float* bufB = bufA + (size_t)n * 64;    // raw aggregated features

  const int B = 256;
  k_fill_one<<<dim3((n + B - 1) / B), dim3(B), 0, stream>>>(dinv, n);
  k_count_deg<<<dim3((e + B - 1) / B), dim3(B), 0, stream>>>(dst, dinv, e);
  k_to_dinv<<<dim3((n + B - 1) / B), dim3(B), 0, stream>>>(dinv, n);

  // layer 1: 8 -> 16 ; layer 2: 16 -> 32 ; layer 3: 32 -> 64
  run_layer<8, 16, false>(x,    W1, nullptr, dinv, src, dst, bufA, bufB, n, e, stream);
  run_layer<16, 32, true>(bufB, W2, b1,      dinv, src, dst, bufA, bufB, n, e, stream);
  run_layer<32, 64, true>(bufB, W3, b2,      dinv, src, dst, bufA, bufB, n, e, stream);

  // final FC: out = relu(h3_raw + b3) @ Wfc + bfc (pre+post fused in WMMA kernel)
  long long gemm_threads = (long long)(n / 16) * 32;
  k_gemm_wmma<64, 64, true, true><<<dim3((gemm_threads + B - 1) / B), dim3(B), 0, stream>>>(
      bufB, Wfc, b3, bfc, (float*)d_out, n);
}